// GIN_Graph_44272522887306
// MI455X (gfx1250) — compile-verified
//
#include <hip/hip_runtime.h>

#define N_NODES   100000
#define N_EDGES   1600000
#define HID       64
#define OUT_C     10
#define N_CONVS   3
#define N_LAYERS  64          // 1 first block + 63 scanned blocks per conv
#define N_GRAPHS  512
#define BN_EPS    1e-5f
#define NSTRIPS   (N_NODES / 16)   // 6250, exact

typedef __attribute__((ext_vector_type(2))) float v2f;
typedef __attribute__((ext_vector_type(8))) float v8f;

// ---------------------------------------------------------------------------
// Fused layer: Z_out = relu(affine_prevBN(Z_in) @ W + b), plus per-channel
// sum / sum-of-squares of Z_out accumulated into stats_out[0..63 | 64..127].
// The previous layer's BN is folded into W'/b' per block (64x64 — cheap).
// Each wave: one 16-row strip, 4 N-tiles, K=64 via 16x V_WMMA_F32_16X16X4_F32.
// ---------------------------------------------------------------------------
__global__ void __launch_bounds__(256)
gemm_relu_stats(const float* __restrict__ Ain, float* __restrict__ Zout,
                const float* __restrict__ Wraw, const float* __restrict__ braw,
                int first_layer,
                const float* __restrict__ prev_stats,
                const float* __restrict__ prev_g, const float* __restrict__ prev_be,
                float* __restrict__ stats_out)
{
    __shared__ float ldsW[4096];   // frag-packed folded weights
    __shared__ float aAff[64], cAff[64], bP[64];
    __shared__ float sSum[64], sSq[64];

    const int tid = threadIdx.x;

    // Per-channel affine of the *previous* BN (identity for first layer).
    if (tid < 64) {
        float a, cc;
        if (first_layer) { a = 1.0f; cc = 0.0f; }
        else {
            float m  = prev_stats[tid]      * (1.0f / (float)N_NODES);
            float q  = prev_stats[64 + tid] * (1.0f / (float)N_NODES);
            float v  = q - m * m;
            float rs = rsqrtf(v + BN_EPS);
            a  = prev_g[tid] * rs;
            cc = prev_be[tid] - m * a;
        }
        aAff[tid] = a; cAff[tid] = cc;
        sSum[tid] = 0.0f; sSq[tid] = 0.0f;
    }
    __syncthreads();

    // Pack W'[k][n] = aAff[k]*W[k][n] into WMMA-B fragment order:
    // frag f = kstep*4 + ntile; lane<16 holds K=4k+0/1, lane>=16 holds K=4k+2/3.
    for (int idx = tid; idx < 4096; idx += 256) {
        int f    = idx >> 6;
        int rem  = idx & 63;
        int lane = rem >> 1;
        int s    = rem & 1;
        int kstep = f >> 2;
        int n     = f & 3;
        int kk  = (kstep << 2) + ((lane >> 4) << 1) + s;
        int col = (n << 4) + (lane & 15);
        ldsW[(f << 6) + (lane << 1) + s] = aAff[kk] * Wraw[kk * 64 + col];
    }
    // b'[n] = b[n] + sum_k cAff[k]*W[k][n]
    if (tid < 64) {
        float acc = braw[tid];
        #pragma unroll 8
        for (int k = 0; k < 64; ++k) acc += cAff[k] * Wraw[k * 64 + tid];
        bP[tid] = acc;
    }
    __syncthreads();

    const int wave = tid >> 5;
    const int lane = tid & 31;
    const int strip = blockIdx.x * 8 + wave;

    if (strip < NSTRIPS) {
        const int row0 = strip << 4;
        const int mrow = row0 + (lane & 15);
        const float* aPtr = Ain + mrow * 64 + ((lane >> 4) << 1);

        v8f acc[4];
        #pragma unroll
        for (int n = 0; n < 4; ++n) {
            float bv = bP[(n << 4) + (lane & 15)];
            #pragma unroll
            for (int r = 0; r < 8; ++r) acc[n][r] = bv;
        }

        #pragma unroll
        for (int k = 0; k < 16; ++k) {
            v2f af = *(const v2f*)(aPtr + k * 4);
            #pragma unroll
            for (int n = 0; n < 4; ++n) {
                v2f bf2 = *(const v2f*)(&ldsW[(((k << 2) + n) << 6) + (lane << 1)]);
                acc[n] = __builtin_amdgcn_wmma_f32_16x16x4_f32(
                    false, af, false, bf2, (short)0, acc[n], false, false);
            }
        }

        // ReLU, store, per-channel stats (LDS atomics, flushed once per block).
        const int mbase = row0 + ((lane >> 4) << 3);
        #pragma unroll
        for (int n = 0; n < 4; ++n) {
            int col = (n << 4) + (lane & 15);
            float s = 0.0f, q = 0.0f;
            #pragma unroll
            for (int r = 0; r < 8; ++r) {
                float z = fmaxf(acc[n][r], 0.0f);
                Zout[(mbase + r) * 64 + col] = z;
                s += z; q += z * z;
            }
            atomicAdd(&sSum[col], s);
            atomicAdd(&sSq[col], q);
        }
    }
    __syncthreads();
    if (tid < 64)       atomicAdd(&stats_out[tid], sSum[tid]);
    else if (tid < 128) atomicAdd(&stats_out[tid], sSq[tid - 64]);
}

// Materialize affine (a, c) of a BN layer from raw stats into aff[0..63|64..127].
__global__ void compute_affine(float* __restrict__ aff, const float* __restrict__ st,
                               const float* __restrict__ g, const float* __restrict__ be)
{
    int t = threadIdx.x;
    float m  = st[t]      * (1.0f / (float)N_NODES);
    float q  = st[64 + t] * (1.0f / (float)N_NODES);
    float rs = rsqrtf(q - m * m + BN_EPS);
    float a  = g[t] * rs;
    aff[t]      = a;
    aff[64 + t] = be[t] - m * a;
}

// H[node] = affine(feat[node])  (self term of GIN, eps=0)
__global__ void node_init(float* __restrict__ H, const float* __restrict__ Zin,
                          int first, const float* __restrict__ aff)
{
    int gid = blockIdx.x * blockDim.x + threadIdx.x;
    if (gid >= N_NODES * 16) return;
    int node = gid >> 4;
    int cg   = (gid & 15) << 2;
    float4 v = *(const float4*)(Zin + node * 64 + cg);
    if (!first) {
        v.x = aff[cg + 0] * v.x + aff[64 + cg + 0];
        v.y = aff[cg + 1] * v.y + aff[64 + cg + 1];
        v.z = aff[cg + 2] * v.z + aff[64 + cg + 2];
        v.w = aff[cg + 3] * v.w + aff[64 + cg + 3];
    }
    *(float4*)(H + node * 64 + cg) = v;
}

// H[dst[e]] += affine(feat[src[e]])
__global__ void edge_scatter(float* __restrict__ H, const float* __restrict__ Zin,
                             const int* __restrict__ src, const int* __restrict__ dst,
                             int first, const float* __restrict__ aff)
{
    int gid = blockIdx.x * blockDim.x + threadIdx.x;
    if (gid >= N_EDGES * 16) return;
    int e  = gid >> 4;
    int cg = (gid & 15) << 2;
    int s = src[e], d = dst[e];
    float4 v = *(const float4*)(Zin + s * 64 + cg);
    float va[4] = {v.x, v.y, v.z, v.w};
    if (!first) {
        #pragma unroll
        for (int j = 0; j < 4; ++j) va[j] = aff[cg + j] * va[j] + aff[64 + cg + j];
    }
    float* hp = H + d * 64 + cg;
    atomicAdd(hp + 0, va[0]);
    atomicAdd(hp + 1, va[1]);
    atomicAdd(hp + 2, va[2]);
    atomicAdd(hp + 3, va[3]);
}

// pooled[batch[node]] += affine(z[node])   (global_add_pool with final BN folded)
__global__ void pool_scatter(float* __restrict__ pooled, const float* __restrict__ Zin,
                             const int* __restrict__ batch, const float* __restrict__ aff)
{
    int gid = blockIdx.x * blockDim.x + threadIdx.x;
    if (gid >= N_NODES * 16) return;
    int node = gid >> 4;
    int cg   = (gid & 15) << 2;
    int g = batch[node];
    float4 v = *(const float4*)(Zin + node * 64 + cg);
    float* pp = pooled + g * 64 + cg;
    atomicAdd(pp + 0, aff[cg + 0] * v.x + aff[64 + cg + 0]);
    atomicAdd(pp + 1, aff[cg + 1] * v.y + aff[64 + cg + 1]);
    atomicAdd(pp + 2, aff[cg + 2] * v.z + aff[64 + cg + 2]);
    atomicAdd(pp + 3, aff[cg + 3] * v.w + aff[64 + cg + 3]);
}

// out = log_softmax(relu(pooled @ W1 + b1) @ Wc + bc); one block per graph.
__global__ void head(float* __restrict__ out, const float* __restrict__ pooled,
                     const float* __restrict__ W1, const float* __restrict__ b1,
                     const float* __restrict__ Wc, const float* __restrict__ bc)
{
    __shared__ float p[64], h[64], lg[OUT_C], red[2];
    int g = blockIdx.x, t = threadIdx.x;
    p[t] = pooled[g * 64 + t];
    __syncthreads();
    float acc = b1[t];
    #pragma unroll 8
    for (int k = 0; k < 64; ++k) acc += p[k] * W1[k * 64 + t];
    h[t] = fmaxf(acc, 0.0f);
    __syncthreads();
    if (t < OUT_C) {
        float a2 = bc[t];
        #pragma unroll 8
        for (int k = 0; k < 64; ++k) a2 += h[k] * Wc[k * OUT_C + t];
        lg[t] = a2;
    }
    __syncthreads();
    if (t == 0) {
        float mx = lg[0];
        for (int i = 1; i < OUT_C; ++i) mx = fmaxf(mx, lg[i]);
        float se = 0.0f;
        for (int i = 0; i < OUT_C; ++i) se += expf(lg[i] - mx);
        red[0] = mx; red[1] = logf(se);
    }
    __syncthreads();
    if (t < OUT_C) out[g * OUT_C + t] = lg[t] - red[0] - red[1];
}

__global__ void zero_f32(float* __restrict__ p, int n)
{
    int i = blockIdx.x * blockDim.x + threadIdx.x;
    if (i < n) p[i] = 0.0f;
}

// ---------------------------------------------------------------------------
extern "C" void kernel_launch(void* const* d_in, const int* in_sizes, int n_in,
                              void* d_out, int out_size, void* d_ws, size_t ws_size,
                              hipStream_t stream)
{
    const float* x    = (const float*)d_in[0];
    // d_in[1] = edge_attr: unused by the reference model
    const int*   ei   = (const int*)d_in[2];
    const int*   src  = ei;
    const int*   dst  = ei + N_EDGES;
    const int*   batch= (const int*)d_in[3];
    const float* Wf   = (const float*)d_in[4];
    const float* bf   = (const float*)d_in[5];
    const float* g0   = (const float*)d_in[6];
    const float* be0  = (const float*)d_in[7];
    const float* Ws   = (const float*)d_in[8];
    const float* bs   = (const float*)d_in[9];
    const float* gs   = (const float*)d_in[10];
    const float* bes  = (const float*)d_in[11];
    const float* W1   = (const float*)d_in[12];
    const float* b1   = (const float*)d_in[13];
    const float* Wc   = (const float*)d_in[14];
    const float* bc   = (const float*)d_in[15];
    float* out = (float*)d_out;

    // Workspace layout (floats): two ping-pong activation buffers (25.6 MB each,
    // both L2-resident), raw BN stats per layer, pooled, affine scratch.
    float* Z0     = (float*)d_ws;
    float* Z1     = Z0 + (size_t)N_NODES * HID;
    float* stats  = Z1 + (size_t)N_NODES * HID;               // 192*128
    float* pooled = stats + N_CONVS * N_LAYERS * 128;         // 512*64
    float* aff    = pooled + N_GRAPHS * HID;                  // 128

    const int nZero = N_CONVS * N_LAYERS * 128 + N_GRAPHS * HID;
    zero_f32<<<(nZero + 255) / 256, 256, 0, stream>>>(stats, nZero);

    float* cur   = nullptr;   // raw z of previous conv's last layer
    float* freeb = Z0;

    for (int c = 0; c < N_CONVS; ++c) {
        float* H = freeb;
        int first = (c == 0);
        const float* feat = first ? x : cur;
        if (!first) {
            compute_affine<<<1, 64, 0, stream>>>(
                aff, stats + ((c - 1) * 64 + 63) * 128,
                gs  + ((c - 1) * 63 + 62) * 64,
                bes + ((c - 1) * 63 + 62) * 64);
        }
        node_init  <<<(N_NODES * 16 + 255) / 256, 256, 0, stream>>>(H, feat, first, aff);
        edge_scatter<<<(N_EDGES * 16 + 255) / 256, 256, 0, stream>>>(H, feat, src, dst, first, aff);

        float* ain  = H;
        float* aout = (H == Z0) ? Z1 : Z0;
        for (int l = 0; l < N_LAYERS; ++l) {
            const float *Wl, *bl, *pst = nullptr, *pg = nullptr, *pb = nullptr;
            int fl = (l == 0);
            if (fl) { Wl = Wf + c * 4096; bl = bf + c * 64; }
            else {
                Wl  = Ws + (c * 63 + (l - 1)) * 4096;
                bl  = bs + (c * 63 + (l - 1)) * 64;
                pst = stats + (c * 64 + (l - 1)) * 128;
                if (l == 1) { pg = g0 + c * 64;               pb = be0 + c * 64; }
                else        { pg = gs + (c * 63 + (l - 2)) * 64; pb = bes + (c * 63 + (l - 2)) * 64; }
            }
            gemm_relu_stats<<<(NSTRIPS + 7) / 8, 256, 0, stream>>>(
                ain, aout, Wl, bl, fl, pst, pg, pb, stats + (c * 64 + l) * 128);
            float* t = ain; ain = aout; aout = t;
        }
        cur   = ain;    // after 64 swaps this is H again (holds final raw z)
        freeb = aout;   // the other buffer is free for the next conv
    }

    compute_affine<<<1, 64, 0, stream>>>(
        aff, stats + (2 * 64 + 63) * 128, gs + (2 * 63 + 62) * 64, bes + (2 * 63 + 62) * 64);
    pool_scatter<<<(N_NODES * 16 + 255) / 256, 256, 0, stream>>>(pooled, cur, batch, aff);
    head<<<N_GRAPHS, 64, 0, stream>>>(out, pooled, W1, b1, Wc, bc);
}